// RGCN_62337155334423
// MI455X (gfx1250) — compile-verified
//
#include <hip/hip_runtime.h>
#include <hip/hip_bf16.h>

typedef __attribute__((ext_vector_type(2))) float v2f;
typedef __attribute__((ext_vector_type(8))) float v8f;

#define NUM_RELS 8
#define BM 128
#define BN 64
#define BK 32
#define A_PITCH 34    // 32 + pad: distinct banks for the 16 rows read per half-wave
#define BP2 130       // B pair-row pitch (floats): BN*2 + 2, even => 8B-aligned rows

// ---------------------------------------------------------------------------
// Degree count: deg[dst*8 + rel] += 1 (hardware fp32 atomic)
// ---------------------------------------------------------------------------
__global__ __launch_bounds__(256) void rgcn_degree_kernel(
    const long long* __restrict__ dst, const long long* __restrict__ rel,
    float* __restrict__ deg, int E) {
  int e = blockIdx.x * blockDim.x + threadIdx.x;
  if (e >= E) return;
  int d = (int)dst[e];
  int r = (int)rel[e];
  unsafeAtomicAdd(&deg[d * NUM_RELS + r], 1.0f);
}

__global__ __launch_bounds__(256) void rgcn_invdeg_kernel(float* __restrict__ deg, int n) {
  int i = blockIdx.x * blockDim.x + threadIdx.x;
  if (i >= n) return;
  deg[i] = 1.0f / fmaxf(deg[i], 1.0f);
}

// ---------------------------------------------------------------------------
// Scatter: one wave per edge; for edges of relation r:
//   tmp[dst, :256] += X[src, :256]   (8 floats / lane, fp32 atomics)
// X (51 MB) is L2-resident, so gathers hit L2 not HBM.
// ---------------------------------------------------------------------------
__global__ __launch_bounds__(256) void rgcn_scatter_kernel(
    const float* __restrict__ X, const long long* __restrict__ src,
    const long long* __restrict__ dst, const long long* __restrict__ rel,
    int r, float* __restrict__ tmp, int E) {
  int gid  = blockIdx.x * blockDim.x + threadIdx.x;
  int e    = gid >> 5;
  int lane = gid & 31;
  if (e >= E) return;
  if ((int)rel[e] != r) return;   // uniform per wave
  long long s = src[e];
  long long d = dst[e];
  const float4* xs = (const float4*)(X + s * 256 + lane * 8);
  float4 v0 = xs[0];
  float4 v1 = xs[1];
  float* t = tmp + d * 256 + lane * 8;
  unsafeAtomicAdd(t + 0, v0.x);
  unsafeAtomicAdd(t + 1, v0.y);
  unsafeAtomicAdd(t + 2, v0.z);
  unsafeAtomicAdd(t + 3, v0.w);
  unsafeAtomicAdd(t + 4, v1.x);
  unsafeAtomicAdd(t + 5, v1.y);
  unsafeAtomicAdd(t + 6, v1.z);
  unsafeAtomicAdd(t + 7, v1.w);
}

// ---------------------------------------------------------------------------
// WMMA fp32 GEMM:  C[M,G] (+)= (A[M,K] * rowscale) @ B[K,G]
//   flags bit0: init (C = bias + result), else C += result (RMW)
//   flags bit1: ReLU the final value
// Block: 256 thr (8 waves); tile 128x64; each wave: 16 rows x 64 cols.
// B is staged K-pair-interleaved in LDS so each B fragment is one ds_load_b64
// landing in an even-aligned VGPR pair (no repack movs before v_wmma).
// ---------------------------------------------------------------------------
__global__ __launch_bounds__(256) void rgcn_gemm_kernel(
    const float* __restrict__ A, int lda,
    const float* __restrict__ scale, int sstride,
    const float* __restrict__ B,
    const float* __restrict__ bias,
    float* __restrict__ C, int ldc,
    int M, int K, int G, int flags) {
  __shared__ float sA[BM * A_PITCH];
  __shared__ float sB2[(BK / 2) * BP2];  // [pairRow][n*2 + j] = B[2*pairRow + j][n]

  const int tid  = threadIdx.x;
  const int wave = tid >> 5;
  const int lane = tid & 31;
  const int half = lane >> 4;   // 0: lanes 0-15, 1: lanes 16-31
  const int l16  = lane & 15;
  const int m0 = blockIdx.x * BM;
  const int n0 = blockIdx.y * BN;

  v8f acc[BN / 16] = {};

  const int arow_t = tid >> 3;          // 0..31
  const int acol   = (tid & 7) * 4;     // 0..28 step 4
  const int bpair  = tid >> 4;          // 0..15 : pair-row (K rows 2p, 2p+1)
  const int bc4    = (tid & 15) * 4;    // 0..60 : column base

  for (int k0 = 0; k0 < K; k0 += BK) {
    // ---- stage A tile (with per-row 1/deg scaling folded in) ----
#pragma unroll
    for (int s = 0; s < 4; ++s) {
      int row  = s * 32 + arow_t;
      int grow = m0 + row;
      float4 v = make_float4(0.f, 0.f, 0.f, 0.f);
      float sc = 1.0f;
      if (grow < M) {
        v = *(const float4*)(A + (long long)grow * lda + k0 + acol);
        if (scale) sc = scale[(long long)grow * sstride];
      }
      float* sa = sA + row * A_PITCH + acol;
      sa[0] = v.x * sc; sa[1] = v.y * sc; sa[2] = v.z * sc; sa[3] = v.w * sc;
    }
    // ---- stage B tile, K-pair interleaved ----
    {
      const float4 r0 = *(const float4*)(B + (long long)(k0 + 2 * bpair)     * G + n0 + bc4);
      const float4 r1 = *(const float4*)(B + (long long)(k0 + 2 * bpair + 1) * G + n0 + bc4);
      float* sb = sB2 + bpair * BP2 + bc4 * 2;
      *(v2f*)(sb + 0) = v2f{r0.x, r1.x};
      *(v2f*)(sb + 2) = v2f{r0.y, r1.y};
      *(v2f*)(sb + 4) = v2f{r0.z, r1.z};
      *(v2f*)(sb + 6) = v2f{r0.w, r1.w};
    }
    // prefetch next K tile (gfx1250 global_prefetch_b8; speculative => safe)
    if (k0 + BK < K) {
      __builtin_prefetch(A + (long long)(m0 + arow_t) * lda + k0 + BK + acol, 0, 0);
      __builtin_prefetch(B + (long long)(k0 + BK + 2 * bpair) * G + n0 + bc4, 0, 0);
    }
    __syncthreads();

    // ---- compute: 8 k-steps of V_WMMA_F32_16X16X4_F32 ----
    const int mrow = wave * 16 + l16;
#pragma unroll
    for (int ks = 0; ks < BK; ks += 4) {
      // A frag: lanes0-15 -> K=ks..ks+1 ; lanes16-31 -> K=ks+2..ks+3
      int kk = ks + half * 2;
      v2f a = *(const v2f*)(sA + mrow * A_PITCH + kk);
      // B pair-row index: {B[kk][n], B[kk+1][n]} contiguous
      const float* brow = sB2 + (ks / 2 + half) * BP2;
#pragma unroll
      for (int ns = 0; ns < BN / 16; ++ns) {
        int n = ns * 16 + l16;
        v2f b = *(const v2f*)(brow + n * 2);
        acc[ns] = __builtin_amdgcn_wmma_f32_16x16x4_f32(
            false, a, false, b, (short)0, acc[ns], false, false);
      }
    }
    __syncthreads();
  }

  // ---- epilogue: C/D layout: VGPR j -> M = half*8 + j, N = l16 ----
#pragma unroll
  for (int ns = 0; ns < BN / 16; ++ns) {
    int col = n0 + ns * 16 + l16;
#pragma unroll
    for (int j = 0; j < 8; ++j) {
      int grow = m0 + wave * 16 + half * 8 + j;
      if (grow < M) {
        float v = acc[ns][j];
        long long idx = (long long)grow * ldc + col;
        if (flags & 1) v += bias[col];
        else           v += C[idx];
        if (flags & 2) v = fmaxf(v, 0.0f);
        C[idx] = v;
      }
    }
  }
}

// ---------------------------------------------------------------------------
extern "C" void kernel_launch(void* const* d_in, const int* in_sizes, int n_in,
                              void* d_out, int out_size, void* d_ws, size_t ws_size,
                              hipStream_t stream) {
  const float*     x     = (const float*)d_in[0];
  const long long* eidx  = (const long long*)d_in[1];  // [2, E] int64
  const long long* etype = (const long long*)d_in[2];  // [E]    int64
  const float*     W1    = (const float*)d_in[3];      // [8,256,256]
  const float*     root1 = (const float*)d_in[4];      // [256,256]
  const float*     b1    = (const float*)d_in[5];      // [256]
  const float*     W2    = (const float*)d_in[6];      // [8,256,128]
  const float*     root2 = (const float*)d_in[7];      // [256,128]
  const float*     b2    = (const float*)d_in[8];      // [128]

  const int N = in_sizes[0] / 256;
  const int E = in_sizes[2];
  const long long* src = eidx;
  const long long* dst = eidx + E;

  // workspace layout (~104 MB): tmp[N*256] | deg[N*8] | h[N*256]
  float* tmp = (float*)d_ws;
  float* deg = tmp + (size_t)N * 256;
  float* h   = deg + (size_t)N * NUM_RELS;
  float* out = (float*)d_out;

  dim3 blk(256);
  const int gm = (N + BM - 1) / BM;
  const int scat_blocks = (int)(((long long)E * 32 + 255) / 256);

  // degrees -> 1/max(deg,1)
  hipMemsetAsync(deg, 0, (size_t)N * NUM_RELS * sizeof(float), stream);
  rgcn_degree_kernel<<<dim3((E + 255) / 256), blk, 0, stream>>>(dst, etype, deg, E);
  rgcn_invdeg_kernel<<<dim3((N * NUM_RELS + 255) / 256), blk, 0, stream>>>(deg, N * NUM_RELS);

  // ---- layer 1: h = relu(b1 + x@root1 + sum_r mean_r(x) @ W1_r) ----
  rgcn_gemm_kernel<<<dim3(gm, 256 / BN), blk, 0, stream>>>(
      x, 256, nullptr, 0, root1, b1, h, 256, N, 256, 256, /*init*/1);
  for (int r = 0; r < NUM_RELS; ++r) {
    hipMemsetAsync(tmp, 0, (size_t)N * 256 * sizeof(float), stream);
    rgcn_scatter_kernel<<<scat_blocks, blk, 0, stream>>>(x, src, dst, etype, r, tmp, E);
    int flags = (r == NUM_RELS - 1) ? 2 : 0;  // ReLU folded into last accumulate
    rgcn_gemm_kernel<<<dim3(gm, 256 / BN), blk, 0, stream>>>(
        tmp, 256, deg + r, NUM_RELS, W1 + (size_t)r * 256 * 256, nullptr,
        h, 256, N, 256, 256, flags);
  }

  // ---- layer 2: out = b2 + h@root2 + sum_r mean_r(h) @ W2_r ----
  rgcn_gemm_kernel<<<dim3(gm, 128 / BN), blk, 0, stream>>>(
      h, 256, nullptr, 0, root2, b2, out, 128, N, 256, 128, /*init*/1);
  for (int r = 0; r < NUM_RELS; ++r) {
    hipMemsetAsync(tmp, 0, (size_t)N * 256 * sizeof(float), stream);
    rgcn_scatter_kernel<<<scat_blocks, blk, 0, stream>>>(h, src, dst, etype, r, tmp, E);
    rgcn_gemm_kernel<<<dim3(gm, 128 / BN), blk, 0, stream>>>(
        tmp, 256, deg + r, NUM_RELS, W2 + (size_t)r * 256 * 128, nullptr,
        out, 128, N, 256, 128, 0);
  }
}